// SimSiamHead_channel_17617955848919
// MI455X (gfx1250) — compile-verified
//
#include <hip/hip_runtime.h>

#define NH 12
#define CB 131072          // elements per (b,head,channel) plane = 512*256
#define PER_BH 393216      // 3*CB
#define NPOS 1048576       // 8 * CB positions per (latent, head)
#define NCHUNK4 8192       // NPOS / 128  (128 positions per wave-chunk)
#define GRID_X 256
#define WPB 8              // waves per block (256 threads, wave32)
#define TOT_WAVES (GRID_X * WPB)
#define BN_EPS 1e-5f
#define COS_EPS2 1e-16f    // (1e-8)^2

typedef __attribute__((ext_vector_type(2))) float v2f;
typedef __attribute__((ext_vector_type(8))) float v8f;

// ---------------- WMMA helper (V_WMMA_F32_16X16X4_F32) ----------------
__device__ __forceinline__ v8f wmma4(v2f a, v2f b, v8f c) {
  return __builtin_amdgcn_wmma_f32_16x16x4_f32(false, a, false, b, (short)0, c,
                                               false, false);
}

// XOR-16 half-wave exchange via DS_SWIZZLE_B32 (group-of-32, SWAPX16 = 0x401F)
__device__ __forceinline__ float swz16(float v) {
  return __int_as_float(__builtin_amdgcn_ds_swizzle(__float_as_int(v), 0x401F));
}

// Load augmented-weight A operand (16x4, rows 0..3 stored, rest zero).
// eff layout per (latent,head): 16 floats row-major [m][k].
// A VGPR layout: lanes 0-15 hold (K0,K1), lanes 16-31 hold (K2,K3), row M = lane&15.
__device__ __forceinline__ v2f load_A(const float* __restrict__ e) {
  int lane = threadIdx.x & 31;
  int m = lane & 15;
  v2f a; a.x = 0.0f; a.y = 0.0f;
  if (m < 4) {
    int o = m * 4 + ((lane >> 4) << 1);
    a.x = e[o]; a.y = e[o + 1];
  }
  return a;
}

// B operand (4x16: K={ch0,ch1,ch2,1}) for tile0 (positions = lanes 0..15 data)
__device__ __forceinline__ v2f make_b_t0(float x0, float x1, float x2) {
  bool hi = (threadIdx.x & 31) >= 16;
  float t2 = swz16(x2);                       // lane 16+n <- lane n's ch2
  v2f b; b.x = hi ? t2 : x0; b.y = hi ? 1.0f : x1;
  return b;
}
// tile1 (positions = lanes 16..31 data)
__device__ __forceinline__ v2f make_b_t1(float x0, float x1, float x2) {
  bool hi = (threadIdx.x & 31) >= 16;
  float s0 = swz16(x0);                       // lane n <- lane 16+n's ch0
  float s1 = swz16(x1);
  v2f b; b.x = hi ? x2 : s0; b.y = hi ? 1.0f : s1;
  return b;
}

// D (channels in d[0..2], lanes 0..15) -> next-stage B, with ReLU
__device__ __forceinline__ v2f next_b_relu(v8f d) {
  bool hi = (threadIdx.x & 31) >= 16;
  float h0 = fmaxf(d[0], 0.0f), h1 = fmaxf(d[1], 0.0f), h2 = fmaxf(d[2], 0.0f);
  float t = swz16(h2);
  v2f b; b.x = hi ? t : h0; b.y = hi ? 1.0f : h1;
  return b;
}
// same, no ReLU (projector output z)
__device__ __forceinline__ v2f next_b_lin(v8f d) {
  bool hi = (threadIdx.x & 31) >= 16;
  float t = swz16(d[2]);
  v2f b; b.x = hi ? t : d[0]; b.y = hi ? 1.0f : d[1];
  return b;
}

// ---------------- reductions ----------------
__device__ __forceinline__ float wave_sum(float v) {
  v += __int_as_float(__builtin_amdgcn_ds_swizzle(__float_as_int(v), 0x401F));
  v += __int_as_float(__builtin_amdgcn_ds_swizzle(__float_as_int(v), 0x201F));
  v += __int_as_float(__builtin_amdgcn_ds_swizzle(__float_as_int(v), 0x101F));
  v += __int_as_float(__builtin_amdgcn_ds_swizzle(__float_as_int(v), 0x081F));
  v += __int_as_float(__builtin_amdgcn_ds_swizzle(__float_as_int(v), 0x041F));
  return v;
}

__device__ __forceinline__ void reduce_stats9(float* s, float* smem,
                                              float* __restrict__ gout) {
  int lane = threadIdx.x & 31, wave = threadIdx.x >> 5;
  for (int t = 0; t < 9; ++t) {
    float v = wave_sum(s[t]);
    if (lane == 0) smem[wave * 9 + t] = v;
  }
  __syncthreads();
  if (threadIdx.x < 9) {
    float v = 0.0f;
    for (int w = 0; w < WPB; ++w) v += smem[w * 9 + threadIdx.x];
    atomicAdd(gout + threadIdx.x, v);
  }
}

__device__ __forceinline__ void accum9(float* s, float h0, float h1, float h2) {
  s[0] += h0; s[1] += h1; s[2] += h2;
  s[3] += h0 * h0; s[4] += h1 * h1; s[5] += h2 * h2;
  s[6] += h0 * h1; s[7] += h0 * h2; s[8] += h1 * h2;
}

// ---------------- init / finalize ----------------
__global__ void init_ws_kernel(float* __restrict__ ws, int n) {
  int i = blockIdx.x * blockDim.x + threadIdx.x;
  if (i < n) ws[i] = 0.0f;
}

__global__ void finalize_kernel(const float* __restrict__ acc, float* __restrict__ out) {
  out[0] = (-0.5f / (float)NPOS) * acc[0];
}

// ---------------- pass 1: raw x moments ----------------
__global__ void pass1_kernel(const float* __restrict__ lat1,
                             const float* __restrict__ lat2,
                             float* __restrict__ stats1) {
  const int head = blockIdx.y, lat = blockIdx.z;
  const float* __restrict__ x = lat ? lat2 : lat1;
  const int lane = threadIdx.x & 31;
  const int wg = blockIdx.x * WPB + (threadIdx.x >> 5);
  float s[9];
#pragma unroll
  for (int k = 0; k < 9; ++k) s[k] = 0.0f;
  for (int i = wg; i < NCHUNK4; i += TOT_WAVES) {
    int p = i * 128 + lane * 4;
    int bb = p >> 17, q = p & (CB - 1);
    const float* base = x + (size_t)(bb * NH + head) * PER_BH + q;
    float4 f0 = *(const float4*)(base);
    float4 f1 = *(const float4*)(base + CB);
    float4 f2 = *(const float4*)(base + 2 * CB);
    float a0[4] = {f0.x, f0.y, f0.z, f0.w};
    float a1[4] = {f1.x, f1.y, f1.z, f1.w};
    float a2[4] = {f2.x, f2.y, f2.z, f2.w};
#pragma unroll
    for (int j = 0; j < 4; ++j) accum9(s, a0[j], a1[j], a2[j]);
  }
  __shared__ float red[WPB * 9];
  reduce_stats9(s, red, stats1 + (lat * NH + head) * 16);
}

// ---------------- pass 2: h1 = relu(conv1'(x)) moments ----------------
__global__ void pass2_kernel(const float* __restrict__ lat1,
                             const float* __restrict__ lat2,
                             const float* __restrict__ eff1,
                             float* __restrict__ stats2) {
  const int head = blockIdx.y, lat = blockIdx.z;
  const float* __restrict__ x = lat ? lat2 : lat1;
  const int lane = threadIdx.x & 31;
  const int wg = blockIdx.x * WPB + (threadIdx.x >> 5);
  const float msk = (lane < 16) ? 1.0f : 0.0f;
  v2f A1 = load_A(eff1 + (lat * NH + head) * 16);
  const v8f c0 = {0, 0, 0, 0, 0, 0, 0, 0};
  float s[9];
#pragma unroll
  for (int k = 0; k < 9; ++k) s[k] = 0.0f;
  for (int i = wg; i < NCHUNK4; i += TOT_WAVES) {
    int p = i * 128 + lane * 4;
    int bb = p >> 17, q = p & (CB - 1);
    const float* base = x + (size_t)(bb * NH + head) * PER_BH + q;
    float4 f0 = *(const float4*)(base);
    float4 f1 = *(const float4*)(base + CB);
    float4 f2 = *(const float4*)(base + 2 * CB);
    float a0[4] = {f0.x, f0.y, f0.z, f0.w};
    float a1[4] = {f1.x, f1.y, f1.z, f1.w};
    float a2[4] = {f2.x, f2.y, f2.z, f2.w};
#pragma unroll
    for (int j = 0; j < 4; ++j) {
#pragma unroll
      for (int tile = 0; tile < 2; ++tile) {
        v2f bB = tile ? make_b_t1(a0[j], a1[j], a2[j])
                      : make_b_t0(a0[j], a1[j], a2[j]);
        v8f d = wmma4(A1, bB, c0);
        accum9(s, fmaxf(d[0], 0.0f) * msk, fmaxf(d[1], 0.0f) * msk,
               fmaxf(d[2], 0.0f) * msk);
      }
    }
  }
  __shared__ float red[WPB * 9];
  reduce_stats9(s, red, stats2 + (lat * NH + head) * 16);
}

// ---------------- pass 3: h2 moments ----------------
__global__ void pass3_kernel(const float* __restrict__ lat1,
                             const float* __restrict__ lat2,
                             const float* __restrict__ eff1,
                             const float* __restrict__ eff2,
                             float* __restrict__ stats3) {
  const int head = blockIdx.y, lat = blockIdx.z;
  const float* __restrict__ x = lat ? lat2 : lat1;
  const int lane = threadIdx.x & 31;
  const int wg = blockIdx.x * WPB + (threadIdx.x >> 5);
  const float msk = (lane < 16) ? 1.0f : 0.0f;
  v2f A1 = load_A(eff1 + (lat * NH + head) * 16);
  v2f A2 = load_A(eff2 + (lat * NH + head) * 16);
  const v8f c0 = {0, 0, 0, 0, 0, 0, 0, 0};
  float s[9];
#pragma unroll
  for (int k = 0; k < 9; ++k) s[k] = 0.0f;
  for (int i = wg; i < NCHUNK4; i += TOT_WAVES) {
    int p = i * 128 + lane * 4;
    int bb = p >> 17, q = p & (CB - 1);
    const float* base = x + (size_t)(bb * NH + head) * PER_BH + q;
    float4 f0 = *(const float4*)(base);
    float4 f1 = *(const float4*)(base + CB);
    float4 f2 = *(const float4*)(base + 2 * CB);
    float a0[4] = {f0.x, f0.y, f0.z, f0.w};
    float a1[4] = {f1.x, f1.y, f1.z, f1.w};
    float a2[4] = {f2.x, f2.y, f2.z, f2.w};
#pragma unroll
    for (int j = 0; j < 4; ++j) {
#pragma unroll
      for (int tile = 0; tile < 2; ++tile) {
        v2f bB = tile ? make_b_t1(a0[j], a1[j], a2[j])
                      : make_b_t0(a0[j], a1[j], a2[j]);
        v8f d1 = wmma4(A1, bB, c0);
        v2f b2 = next_b_relu(d1);
        v8f d2 = wmma4(A2, b2, c0);
        accum9(s, fmaxf(d2[0], 0.0f) * msk, fmaxf(d2[1], 0.0f) * msk,
               fmaxf(d2[2], 0.0f) * msk);
      }
    }
  }
  __shared__ float red[WPB * 9];
  reduce_stats9(s, red, stats3 + (lat * NH + head) * 16);
}

// ---------------- pass 4: full chain + cosine loss ----------------
__device__ __forceinline__ void chain5(v2f bx, v2f A1, v2f A2, v2f A3, v2f A4,
                                       v2f A5, float z[3], float pr[3]) {
  const bool hi = (threadIdx.x & 31) >= 16;
  const v8f c0 = {0, 0, 0, 0, 0, 0, 0, 0};
  v8f d1 = wmma4(A1, bx, c0);
  v2f b2 = next_b_relu(d1);
  v8f d2 = wmma4(A2, b2, c0);
  v2f b3 = next_b_relu(d2);
  v8f d3 = wmma4(A3, b3, c0);                 // z (no relu)
  z[0] = d3[0]; z[1] = d3[1]; z[2] = d3[2];
  v2f bz = next_b_lin(d3);
  v8f d4 = wmma4(A4, bz, c0);                 // s4 (1 channel, row 0)
  float h4 = fmaxf(d4[0], 0.0f);
  v2f b5; b5.x = hi ? 0.0f : h4; b5.y = hi ? 0.0f : 1.0f;
  v8f d5 = wmma4(A5, b5, c0);                 // p
  pr[0] = d5[0]; pr[1] = d5[1]; pr[2] = d5[2];
}

__device__ __forceinline__ float cos_pair(const float* z1, const float* p1,
                                          const float* z2, const float* p2) {
  float d12 = p1[0]*z2[0] + p1[1]*z2[1] + p1[2]*z2[2];
  float pp1 = p1[0]*p1[0] + p1[1]*p1[1] + p1[2]*p1[2];
  float zz2 = z2[0]*z2[0] + z2[1]*z2[1] + z2[2]*z2[2];
  float d21 = p2[0]*z1[0] + p2[1]*z1[1] + p2[2]*z1[2];
  float pp2 = p2[0]*p2[0] + p2[1]*p2[1] + p2[2]*p2[2];
  float zz1 = z1[0]*z1[0] + z1[1]*z1[1] + z1[2]*z1[2];
  // max(sqrt(x), eps) == sqrt(max(x, eps^2)) -> use rsqrt, no divides
  float t12 = d12 * rsqrtf(fmaxf(pp1, COS_EPS2)) * rsqrtf(fmaxf(zz2, COS_EPS2));
  float t21 = d21 * rsqrtf(fmaxf(pp2, COS_EPS2)) * rsqrtf(fmaxf(zz1, COS_EPS2));
  return t12 + t21;
}

__global__ void pass4_kernel(const float* __restrict__ lat1,
                             const float* __restrict__ lat2,
                             const float* __restrict__ eff1,
                             const float* __restrict__ eff2,
                             const float* __restrict__ eff3,
                             const float* __restrict__ eff4,
                             const float* __restrict__ eff5,
                             float* __restrict__ acc_out) {
  const int head = blockIdx.y;
  const int lane = threadIdx.x & 31;
  const int wg = blockIdx.x * WPB + (threadIdx.x >> 5);
  const float msk = (lane < 16) ? 1.0f : 0.0f;
  v2f A1[2], A2[2], A3[2], A4[2], A5[2];
#pragma unroll
  for (int l = 0; l < 2; ++l) {
    int o = (l * NH + head) * 16;
    A1[l] = load_A(eff1 + o); A2[l] = load_A(eff2 + o); A3[l] = load_A(eff3 + o);
    A4[l] = load_A(eff4 + o); A5[l] = load_A(eff5 + o);
  }
  float acc = 0.0f;
  for (int i = wg; i < NCHUNK4; i += TOT_WAVES) {
    int p = i * 128 + lane * 4;
    int bb = p >> 17, q = p & (CB - 1);
    size_t base = (size_t)(bb * NH + head) * PER_BH + q;
    float4 u0 = *(const float4*)(lat1 + base);
    float4 u1 = *(const float4*)(lat1 + base + CB);
    float4 u2 = *(const float4*)(lat1 + base + 2 * CB);
    float4 v0 = *(const float4*)(lat2 + base);
    float4 v1 = *(const float4*)(lat2 + base + CB);
    float4 v2 = *(const float4*)(lat2 + base + 2 * CB);
    float y0[4] = {u0.x, u0.y, u0.z, u0.w};
    float y1[4] = {u1.x, u1.y, u1.z, u1.w};
    float y2[4] = {u2.x, u2.y, u2.z, u2.w};
    float w0[4] = {v0.x, v0.y, v0.z, v0.w};
    float w1[4] = {v1.x, v1.y, v1.z, v1.w};
    float w2[4] = {v2.x, v2.y, v2.z, v2.w};
#pragma unroll
    for (int j = 0; j < 4; ++j) {
#pragma unroll
      for (int tile = 0; tile < 2; ++tile) {
        v2f bA = tile ? make_b_t1(y0[j], y1[j], y2[j])
                      : make_b_t0(y0[j], y1[j], y2[j]);
        v2f bBv = tile ? make_b_t1(w0[j], w1[j], w2[j])
                       : make_b_t0(w0[j], w1[j], w2[j]);
        float z1[3], p1[3], z2[3], p2[3];
        chain5(bA, A1[0], A2[0], A3[0], A4[0], A5[0], z1, p1);
        chain5(bBv, A1[1], A2[1], A3[1], A4[1], A5[1], z2, p2);
        acc += msk * cos_pair(z1, p1, z2, p2);
      }
    }
  }
  float v = wave_sum(acc);
  __shared__ float red[WPB];
  if (lane == 0) red[threadIdx.x >> 5] = v;
  __syncthreads();
  if (threadIdx.x == 0) {
    float t = 0.0f;
    for (int w = 0; w < WPB; ++w) t += red[w];
    atomicAdd(acc_out, t);
  }
}

// ---------------- solve kernels (fold BN into effective conv) ----------------
__global__ void solve_bn3x3_kernel(const float* __restrict__ stats,
                                   const float* __restrict__ W,
                                   const float* __restrict__ b,
                                   const float* __restrict__ g,
                                   const float* __restrict__ be,
                                   float* __restrict__ eff) {
  int t = threadIdx.x; if (t >= 24) return;
  int h = t % NH;
  const float* S = stats + t * 16;
  const float invP = 1.0f / (float)NPOS;
  float m0 = S[0] * invP, m1 = S[1] * invP, m2 = S[2] * invP;
  float Cm[3][3];
  Cm[0][0] = S[3] * invP - m0 * m0; Cm[1][1] = S[4] * invP - m1 * m1;
  Cm[2][2] = S[5] * invP - m2 * m2;
  Cm[0][1] = Cm[1][0] = S[6] * invP - m0 * m1;
  Cm[0][2] = Cm[2][0] = S[7] * invP - m0 * m2;
  Cm[1][2] = Cm[2][1] = S[8] * invP - m1 * m2;
  float* e = eff + t * 16;
  for (int o = 0; o < 3; ++o) {
    const float* w = W + (h * 3 + o) * 3;
    float mean = w[0] * m0 + w[1] * m1 + w[2] * m2 + b[h * 3 + o];
    float var = 0.0f;
    for (int k = 0; k < 3; ++k)
      for (int j = 0; j < 3; ++j) var += w[k] * Cm[k][j] * w[j];
    float a = g[h * 3 + o] * rsqrtf(var + BN_EPS);
    e[o * 4 + 0] = a * w[0]; e[o * 4 + 1] = a * w[1]; e[o * 4 + 2] = a * w[2];
    e[o * 4 + 3] = a * (b[h * 3 + o] - mean) + be[h * 3 + o];
  }
}

__global__ void solve_stage3_kernel(const float* __restrict__ stats3,
    const float* __restrict__ W3, const float* __restrict__ b3,
    const float* __restrict__ g3, const float* __restrict__ be3,
    const float* __restrict__ W4, const float* __restrict__ b4,
    const float* __restrict__ g4, const float* __restrict__ be4,
    const float* __restrict__ W5, const float* __restrict__ b5,
    float* __restrict__ eff3, float* __restrict__ eff4,
    float* __restrict__ eff5) {
  int t = threadIdx.x; if (t >= 24) return;
  int h = t % NH;
  const float* S = stats3 + t * 16;
  const float invP = 1.0f / (float)NPOS;
  float m[3] = {S[0] * invP, S[1] * invP, S[2] * invP};
  float Cm[3][3];
  Cm[0][0] = S[3] * invP - m[0] * m[0]; Cm[1][1] = S[4] * invP - m[1] * m[1];
  Cm[2][2] = S[5] * invP - m[2] * m[2];
  Cm[0][1] = Cm[1][0] = S[6] * invP - m[0] * m[1];
  Cm[0][2] = Cm[2][0] = S[7] * invP - m[0] * m[2];
  Cm[1][2] = Cm[2][1] = S[8] * invP - m[1] * m[2];
  // BN3 affine folded into conv3
  float Wp[3][3], bp[3];
  for (int o = 0; o < 3; ++o) {
    const float* w = W3 + (h * 3 + o) * 3;
    float mean = w[0] * m[0] + w[1] * m[1] + w[2] * m[2] + b3[h * 3 + o];
    float var = 0.0f;
    for (int k = 0; k < 3; ++k)
      for (int j = 0; j < 3; ++j) var += w[k] * Cm[k][j] * w[j];
    float a = g3[h * 3 + o] * rsqrtf(var + BN_EPS);
    Wp[o][0] = a * w[0]; Wp[o][1] = a * w[1]; Wp[o][2] = a * w[2];
    bp[o] = a * (b3[h * 3 + o] - mean) + be3[h * 3 + o];
  }
  float* e3 = eff3 + t * 16;
  for (int o = 0; o < 3; ++o) {
    e3[o * 4 + 0] = Wp[o][0]; e3[o * 4 + 1] = Wp[o][1];
    e3[o * 4 + 2] = Wp[o][2]; e3[o * 4 + 3] = bp[o];
  }
  // z statistics (z = conv3'(h2) is linear in h2): mz, Cov(z)
  float mz[3], Cz[3][3];
  for (int o = 0; o < 3; ++o)
    mz[o] = Wp[o][0] * m[0] + Wp[o][1] * m[1] + Wp[o][2] * m[2] + bp[o];
  for (int o = 0; o < 3; ++o)
    for (int p = 0; p < 3; ++p) {
      float s = 0.0f;
      for (int k = 0; k < 3; ++k)
        for (int j = 0; j < 3; ++j) s += Wp[o][k] * Cm[k][j] * Wp[p][j];
      Cz[o][p] = s;
    }
  // BN4 on y4 = W4·z + b4 (1 output channel), folded into conv4
  const float* u = W4 + h * 3;
  float mean4 = u[0] * mz[0] + u[1] * mz[1] + u[2] * mz[2] + b4[h];
  float var4 = 0.0f;
  for (int k = 0; k < 3; ++k)
    for (int j = 0; j < 3; ++j) var4 += u[k] * Cz[k][j] * u[j];
  float a4 = g4[h] * rsqrtf(var4 + BN_EPS);
  float* e4 = eff4 + t * 16;
  e4[0] = a4 * u[0]; e4[1] = a4 * u[1]; e4[2] = a4 * u[2];
  e4[3] = a4 * (b4[h] - mean4) + be4[h];
  // conv5 (no BN): p = W5*h4 + b5 ; B is {h4, 1, 0, 0}
  float* e5 = eff5 + t * 16;
  for (int o = 0; o < 3; ++o) {
    e5[o * 4 + 0] = W5[h * 3 + o]; e5[o * 4 + 1] = b5[h * 3 + o];
  }
}

// ---------------- workspace layout (floats) ----------------
#define WS_ACC 0
#define WS_S1 16
#define WS_S2 400
#define WS_S3 784
#define WS_E1 1168
#define WS_E2 1552
#define WS_E3 1936
#define WS_E4 2320
#define WS_E5 2704
#define WS_TOTAL 3088

extern "C" void kernel_launch(void* const* d_in, const int* in_sizes, int n_in,
                              void* d_out, int out_size, void* d_ws, size_t ws_size,
                              hipStream_t stream) {
  const float* lat1 = (const float*)d_in[0];
  const float* lat2 = (const float*)d_in[1];
  const float* W1 = (const float*)d_in[2];  const float* b1 = (const float*)d_in[3];
  const float* g1 = (const float*)d_in[4];  const float* be1 = (const float*)d_in[5];
  const float* W2 = (const float*)d_in[6];  const float* b2 = (const float*)d_in[7];
  const float* g2 = (const float*)d_in[8];  const float* be2 = (const float*)d_in[9];
  const float* W3 = (const float*)d_in[10]; const float* b3 = (const float*)d_in[11];
  const float* g3 = (const float*)d_in[12]; const float* be3 = (const float*)d_in[13];
  const float* W4 = (const float*)d_in[14]; const float* b4 = (const float*)d_in[15];
  const float* g4 = (const float*)d_in[16]; const float* be4 = (const float*)d_in[17];
  const float* W5 = (const float*)d_in[18]; const float* b5 = (const float*)d_in[19];
  float* ws = (float*)d_ws;
  float* out = (float*)d_out;

  init_ws_kernel<<<(WS_TOTAL + 255) / 256, 256, 0, stream>>>(ws, WS_TOTAL);

  pass1_kernel<<<dim3(GRID_X, NH, 2), 256, 0, stream>>>(lat1, lat2, ws + WS_S1);
  solve_bn3x3_kernel<<<1, 32, 0, stream>>>(ws + WS_S1, W1, b1, g1, be1, ws + WS_E1);

  pass2_kernel<<<dim3(GRID_X, NH, 2), 256, 0, stream>>>(lat1, lat2, ws + WS_E1,
                                                        ws + WS_S2);
  solve_bn3x3_kernel<<<1, 32, 0, stream>>>(ws + WS_S2, W2, b2, g2, be2, ws + WS_E2);

  pass3_kernel<<<dim3(GRID_X, NH, 2), 256, 0, stream>>>(lat1, lat2, ws + WS_E1,
                                                        ws + WS_E2, ws + WS_S3);
  solve_stage3_kernel<<<1, 32, 0, stream>>>(ws + WS_S3, W3, b3, g3, be3,
                                            W4, b4, g4, be4, W5, b5,
                                            ws + WS_E3, ws + WS_E4, ws + WS_E5);

  pass4_kernel<<<dim3(GRID_X, NH, 1), 256, 0, stream>>>(
      lat1, lat2, ws + WS_E1, ws + WS_E2, ws + WS_E3, ws + WS_E4, ws + WS_E5,
      ws + WS_ACC);

  finalize_kernel<<<1, 1, 0, stream>>>(ws + WS_ACC, out);
}